// GraphQNetwork_73847667687761
// MI455X (gfx1250) — compile-verified
//
#include <hip/hip_runtime.h>
#include <math.h>

// ===========================================================================
// GraphQNetwork forward for MI455X (gfx1250), fp32 throughout.
// Dense layers use V_WMMA_F32_16X16X4_F32 (CDNA5 fp32 WMMA, exact fp32 math).
// Weights are pre-packed (zero-padded K->mult-of-4, cols->mult-of-16) so the
// WMMA K-loop has ZERO bounds checks: no exec-mask branches, just loads+wmma.
// Activations get +16 slack rows so tile tails read in-bounds garbage that
// only lands in never-stored accumulator rows.
//
// Input flattening assumption (setup_inputs() dict insertion order, recursive):
//  0 x[N,7] 1 edge_index[2,E] 2 edge_attr[E,1] 3 batch[N] 4 action_mask[N]
//  5 goal[G,6] 6 visited_subgraph[S] 7 subgraph_node_indices_batch[S]
//  8 current_node[G]
//  9-11  norm0.{w,b,ms}[7]      12-14 norm_edge.{w,b,ms}[1]
//  15-21 conv1 {Wl,bl,Wr,br,We,att,bias}   (16 -> 2x5)
//  22-28 conv2 (26 -> 2x10)   29-35 conv3 (46 -> 1x30)  36-42 conv_whole (57 -> 2x22)
//  43-44 merge {W[102,32], b[32]}
//  45-50 graph_embedder {W0[32,16],b0,W1[16,10],b1,W2[10,6],b2}
//  51-56 subgraph_embedder (same shapes)
//  57-60 goal_branches {W1[6,76,32],b1[6,32],W2[6,32,26],b2[6,26]}
//  61-66 qvalue_branches {W1[6,44,32],b1,W2[6,32,16],b2,W3[6,16,1],b3}
// Output: masked q-values [N] float32.
// ===========================================================================

typedef float v2f __attribute__((ext_vector_type(2)));
typedef float v8f __attribute__((ext_vector_type(8)));

#define TPB 256

// Order-invariant atomic float max (exact): positive path via signed int max,
// negative path via unsigned int min.
static __device__ __forceinline__ void atomicMaxF(float* addr, float val) {
  unsigned u = __float_as_uint(val);
  if ((u >> 31) == 0) atomicMax((int*)addr, (int)u);
  else                atomicMin((unsigned int*)addr, u);
}

// ------------------------- GraphNorm statistics ----------------------------
__global__ void k_node_stats1(const float* __restrict__ x, const int* __restrict__ batch,
                              float* nodeSum, float* nodeCnt, int N) {
  int i = blockIdx.x * blockDim.x + threadIdx.x;
  if (i >= N) return;
  int g = batch[i];
#pragma unroll
  for (int f = 0; f < 7; ++f) atomicAdd(&nodeSum[g * 7 + f], x[(size_t)i * 7 + f]);
  atomicAdd(&nodeCnt[g], 1.0f);
}

__global__ void k_edge_stats1(const float* __restrict__ ea, const int* __restrict__ src,
                              const int* __restrict__ batch, float* edgeSum, float* edgeCnt, int E) {
  int e = blockIdx.x * blockDim.x + threadIdx.x;
  if (e >= E) return;
  int g = batch[src[e]];
  atomicAdd(&edgeSum[g], ea[e]);
  atomicAdd(&edgeCnt[g], 1.0f);
}

__global__ void k_means(const float* nodeSum, const float* nodeCnt,
                        const float* edgeSum, const float* edgeCnt,
                        float* nodeMean, float* edgeMean, int G) {
  int g = blockIdx.x * blockDim.x + threadIdx.x;
  if (g >= G) return;
  float cn = fmaxf(nodeCnt[g], 1.0f);
  for (int f = 0; f < 7; ++f) nodeMean[g * 7 + f] = nodeSum[g * 7 + f] / cn;
  float ce = fmaxf(edgeCnt[g], 1.0f);
  edgeMean[g] = edgeSum[g] / ce;
}

__global__ void k_node_stats2(const float* __restrict__ x, const int* __restrict__ batch,
                              const float* nodeMean, const float* ms, float* nodeVar, int N) {
  int i = blockIdx.x * blockDim.x + threadIdx.x;
  if (i >= N) return;
  int g = batch[i];
#pragma unroll
  for (int f = 0; f < 7; ++f) {
    float vc = x[(size_t)i * 7 + f] - nodeMean[g * 7 + f] * ms[f];
    atomicAdd(&nodeVar[g * 7 + f], vc * vc);
  }
}

__global__ void k_edge_stats2(const float* __restrict__ ea, const int* __restrict__ src,
                              const int* __restrict__ batch, const float* edgeMean,
                              const float* ems, float* edgeVar, int E) {
  int e = blockIdx.x * blockDim.x + threadIdx.x;
  if (e >= E) return;
  int g = batch[src[e]];
  float vc = ea[e] - edgeMean[g] * ems[0];
  atomicAdd(&edgeVar[g], vc * vc);
}

__global__ void k_inv(const float* nodeVar, const float* nodeCnt,
                      const float* edgeVar, const float* edgeCnt,
                      float* nodeInv, float* edgeInv, int G) {
  int g = blockIdx.x * blockDim.x + threadIdx.x;
  if (g >= G) return;
  float cn = fmaxf(nodeCnt[g], 1.0f);
  for (int f = 0; f < 7; ++f)
    nodeInv[g * 7 + f] = 1.0f / sqrtf(nodeVar[g * 7 + f] / cn + 1e-5f);
  float ce = fmaxf(edgeCnt[g], 1.0f);
  edgeInv[g] = 1.0f / sqrtf(edgeVar[g] / ce + 1e-5f);
}

// normalized edge attrs + segment means over src (ein) and dst (eout)
__global__ void k_ea(const float* __restrict__ edge_attr, const int* __restrict__ src,
                     const int* __restrict__ dst, const int* __restrict__ batch,
                     const float* edgeMean, const float* edgeInv,
                     const float* ew, const float* eb, const float* ems,
                     float* EA, float* einSum, float* einCnt, float* eoSum, float* eoCnt, int E) {
  int e = blockIdx.x * blockDim.x + threadIdx.x;
  if (e >= E) return;
  int s = src[e], d = dst[e];
  int g = batch[s];
  float vc = edge_attr[e] - edgeMean[g] * ems[0];
  float v = ew[0] * vc * edgeInv[g] + eb[0];
  EA[e] = v;
  atomicAdd(&einSum[s], v); atomicAdd(&einCnt[s], 1.0f);
  atomicAdd(&eoSum[d], v);  atomicAdd(&eoCnt[d], 1.0f);
}

// h16 cols 0-6 = x0 (GraphNorm), col7 = ein, col8 = eout; also EOUT buffer
__global__ void k_h16a(const float* __restrict__ x, const int* __restrict__ batch,
                       const float* nodeMean, const float* nodeInv,
                       const float* nw, const float* nb, const float* nms,
                       const float* einSum, const float* einCnt,
                       const float* eoSum, const float* eoCnt,
                       float* H16, float* EOUT, int N) {
  int i = blockIdx.x * blockDim.x + threadIdx.x;
  if (i >= N) return;
  int g = batch[i];
#pragma unroll
  for (int f = 0; f < 7; ++f)
    H16[(size_t)i * 16 + f] =
        nw[f] * (x[(size_t)i * 7 + f] - nodeMean[g * 7 + f] * nms[f]) * nodeInv[g * 7 + f] + nb[f];
  H16[(size_t)i * 16 + 7] = einSum[i] / fmaxf(einCnt[i], 1.0f);
  float eo = eoSum[i] / fmaxf(eoCnt[i], 1.0f);
  H16[(size_t)i * 16 + 8] = eo;
  EOUT[i] = eo;
}

__global__ void k_cur_gidx(const float* __restrict__ H16, const int* __restrict__ curnode,
                           const float* __restrict__ goal, float* cur, int* gidx, int G) {
  int g = blockIdx.x * blockDim.x + threadIdx.x;
  if (g >= G) return;
  int nd = curnode[g];
#pragma unroll
  for (int f = 0; f < 7; ++f) cur[g * 7 + f] = H16[(size_t)nd * 16 + f];
  float best = goal[g * 6]; int bi = 0;
#pragma unroll
  for (int j = 1; j < 6; ++j) { float v = goal[g * 6 + j]; if (v > best) { best = v; bi = j; } }
  gidx[g] = bi;
}

__global__ void k_ranges(const int* __restrict__ batch, int* gstart, int N, int G) {
  int i = blockIdx.x * blockDim.x + threadIdx.x;
  if (i >= N) return;
  int g = batch[i];
  if (i == 0 || batch[i - 1] != g) gstart[g] = i;
  if (i == 0) gstart[G] = N;
}

__global__ void k_h16b(const float* __restrict__ cur, const int* __restrict__ batch,
                       float* H16, int N) {
  int i = blockIdx.x * blockDim.x + threadIdx.x;
  if (i >= N) return;
  int g = batch[i];
#pragma unroll
  for (int f = 0; f < 7; ++f) H16[(size_t)i * 16 + 9 + f] = cur[g * 7 + f];
}

// ------------------------- weight packing ----------------------------------
// Wp[set][k][n] (K4 x fo16, zero-padded), bp[set][n] (fo16, zero-padded).
__global__ void k_packW(const float* __restrict__ W, const float* __restrict__ b,
                        float* Wp, float* bp, int fin, int fo, int K4, int fo16, int nb) {
  int i = blockIdx.x * blockDim.x + threadIdx.x;
  int tot = nb * K4 * fo16;
  if (i < tot) {
    int n = i % fo16;
    int k = (i / fo16) % K4;
    int s = i / (fo16 * K4);
    Wp[i] = (k < fin && n < fo) ? W[((size_t)s * fin + k) * fo + n] : 0.0f;
  }
  if (i < nb * fo16) {
    int n = i % fo16;
    int s = i / fo16;
    bp[i] = (n < fo) ? b[s * fo + n] : 0.0f;
  }
}

// ------------------------------- WMMA GEMM ---------------------------------
// out[m,n] = act( sum_k in(m,k) * Wp[k,n] + bp[n] ), K stepped by 4 via
// V_WMMA_F32_16X16X4_F32. Guard-free inner loop: Wp/bp are zero-padded, input
// buffers carry +16 slack rows, so all loads are unconditional.
// A 16x4 layout: M=lane&15; VGPR0 K={0|2}, VGPR1 K={1|3} (split at lane 16).
// B 4x16 layout: N=lane&15; same K split.  C/D: elem r -> M=r(+8), N=lane&15.
template <int SPLIT>
static __device__ __forceinline__ void wmma_tile(
    const float* in1, int s1, int f1, const float* in2, int s2, int K4,
    const float* __restrict__ Wp, const float* __restrict__ bp,
    float* out, int so, int fo, int fo16,
    int row0, int rowEnd, int col0, int relu) {
  int l = threadIdx.x & 31;
  int mrow = row0 + (l & 15);
  int khalf = (l >> 4) << 1;   // 0 for lanes 0-15, 2 for lanes 16-31
  int ncol = col0 + (l & 15);
  const float* a1 = in1 + (size_t)mrow * s1;                 // index with k
  const float* a2 = SPLIT ? (in2 + (size_t)mrow * s2 - f1) : a1;  // index with k (k>=f1)
  const float* bw = Wp + ncol;                                // index with k*fo16
  v8f acc = {0.f, 0.f, 0.f, 0.f, 0.f, 0.f, 0.f, 0.f};
  for (int k0 = 0; k0 < K4; k0 += 4) {
    int ka = k0 + khalf;
    const float* pa = SPLIT ? ((ka < f1) ? a1 : a2) : a1;
    v2f a, b;
    a.x = pa[ka];
    a.y = pa[ka + 1];
    b.x = bw[(size_t)ka * fo16];
    b.y = bw[(size_t)ka * fo16 + fo16];
    acc = __builtin_amdgcn_wmma_f32_16x16x4_f32(false, a, false, b, (short)0, acc,
                                                false, false);
  }
  if (ncol < fo) {
    float bv = bp[ncol];
    int rbase = row0 + ((l >> 4) << 3);   // +0 or +8
#pragma unroll
    for (int r = 0; r < 8; ++r) {
      int row = rbase + r;
      if (row < rowEnd) {
        float v = acc[r] + bv;
        if (relu) v = v > 0.0f ? v : 0.0f;
        out[(size_t)row * so + ncol] = v;
      }
    }
  }
}

template <int SPLIT>
__global__ __launch_bounds__(32) void k_gemm(
    const float* in1, int s1, int f1, const float* in2, int s2, int K4,
    const float* Wp, const float* bp, float* out, int so, int fo, int fo16,
    int nrows, int relu) {
  int colT = fo16 >> 4;
  int t = blockIdx.x;
  int rt = t / colT, ct = t % colT;
  int row0 = rt * 16;
  if (row0 >= nrows) return;
  int re = row0 + 16; if (re > nrows) re = nrows;
  wmma_tile<SPLIT>(in1, s1, f1, in2, s2, K4, Wp, bp, out, so, fo, fo16,
                   row0, re, ct * 16, relu);
}

// Per-graph branch GEMM: weights selected by gidx[g]; batch is sorted so each
// graph occupies contiguous rows [gstart[g], gstart[g+1]).
__global__ __launch_bounds__(32) void k_gemm_branch(
    const float* in, int s1, int K4, const float* Wp, const float* bp,
    float* out, int so, int fo, int fo16, const int* gstart, const int* gidx,
    int TPG, int relu) {
  int colT = fo16 >> 4;
  int b = blockIdx.x;
  int g = b / (colT * TPG);
  int rem = b % (colT * TPG);
  int ct = rem / TPG;
  int t = rem % TPG;
  int r0 = gstart[g] + t * 16;
  int rEnd = gstart[g + 1];
  if (r0 >= rEnd) return;
  const float* W = Wp + (size_t)gidx[g] * K4 * fo16;
  const float* bias = bp + (size_t)gidx[g] * fo16;
  int re = r0 + 16; if (re > rEnd) re = rEnd;
  wmma_tile<0>(in, s1, K4, nullptr, 0, K4, W, bias, out, so, fo, fo16,
               r0, re, ct * 16, relu);
}

// ----------------------------- GATv2 edge passes ---------------------------
__global__ void k_fill_att(float* AMAX, float* ASUM, int n) {
  int i = blockIdx.x * blockDim.x + threadIdx.x;
  if (i >= n) return;
  AMAX[i] = -1e38f; ASUM[i] = 0.0f;
}

__global__ void k_edge1(const int* __restrict__ src, const int* __restrict__ dst,
                        const float* __restrict__ EA, const float* __restrict__ EOUT,
                        const float* __restrict__ XL, const float* __restrict__ XR,
                        const float* __restrict__ We, const float* __restrict__ att,
                        float* ABUF, float* AMAX, int H, int C, int E, int Ef) {
  int e = blockIdx.x * blockDim.x + threadIdx.x;
  if (e >= Ef) return;
  int s, d; float w;
  if (e < E) { s = src[e]; d = dst[e]; w = EA[e]; }
  else       { s = d = e - E; w = EOUT[e - E]; }
  int HC = H * C;
  for (int h = 0; h < H; ++h) {
    float acc = 0.0f;
    for (int c = 0; c < C; ++c) {
      int j = h * C + c;
      float v = XL[(size_t)s * HC + j] + XR[(size_t)d * HC + j] + w * We[j];
      v = v > 0.0f ? v : 0.2f * v;      // leaky_relu, slope 0.2
      acc += v * att[j];
    }
    ABUF[(size_t)e * H + h] = acc;
    atomicMaxF(&AMAX[(size_t)d * H + h], acc);
  }
}

__global__ void k_edge2(const int* __restrict__ dst, float* ABUF,
                        const float* __restrict__ AMAX, float* ASUM,
                        int H, int E, int Ef) {
  int e = blockIdx.x * blockDim.x + threadIdx.x;
  if (e >= Ef) return;
  int d = (e < E) ? dst[e] : (e - E);
  for (int h = 0; h < H; ++h) {
    float ae = expf(ABUF[(size_t)e * H + h] - AMAX[(size_t)d * H + h]);
    ABUF[(size_t)e * H + h] = ae;
    atomicAdd(&ASUM[(size_t)d * H + h], ae);
  }
}

__global__ void k_edge3(const int* __restrict__ src, const int* __restrict__ dst,
                        const float* __restrict__ ABUF, const float* __restrict__ ASUM,
                        const float* __restrict__ XL, float* dest, int so,
                        int H, int C, int E, int Ef) {
  int e = blockIdx.x * blockDim.x + threadIdx.x;
  if (e >= Ef) return;
  int s, d;
  if (e < E) { s = src[e]; d = dst[e]; } else { s = d = e - E; }
  int HC = H * C;
  for (int h = 0; h < H; ++h) {
    float alpha = ABUF[(size_t)e * H + h] / (ASUM[(size_t)d * H + h] + 1e-16f);
    for (int c = 0; c < C; ++c) {
      int j = h * C + c;
      atomicAdd(&dest[(size_t)d * so + j], XL[(size_t)s * HC + j] * alpha);
    }
  }
}

// dest[:, :HC] += bias ; dest[:, HC:HC+identLen] = ident  (skip connection)
__global__ void k_pack(float* dest, int so, int HC, const float* __restrict__ bias,
                       const float* __restrict__ ident, int si, int identLen, int N) {
  int i = blockIdx.x * blockDim.x + threadIdx.x;
  if (i >= N) return;
  for (int c = 0; c < HC; ++c) dest[(size_t)i * so + c] += bias[c];
  for (int j = 0; j < identLen; ++j)
    dest[(size_t)i * so + HC + j] = ident[(size_t)i * si + j];
}

// ------------------------------- pooling / MLPs ----------------------------
__global__ void k_fill_neg(float* p, int n) {
  int i = blockIdx.x * blockDim.x + threadIdx.x;
  if (i >= n) return;
  p[i] = -1e38f;
}

__global__ void k_gmax(const float* __restrict__ XG, const int* __restrict__ batch,
                       float* gmax, int N) {
  int i = blockIdx.x * blockDim.x + threadIdx.x;
  if (i >= N) return;
  int g = batch[i];
  for (int c = 0; c < 26; ++c) atomicMaxF(&gmax[g * 26 + c], XG[(size_t)i * 26 + c]);
}

__global__ void k_smax(const float* __restrict__ XG, const int* __restrict__ vsub,
                       const int* __restrict__ subb, float* smax, int S) {
  int s = blockIdx.x * blockDim.x + threadIdx.x;
  if (s >= S) return;
  int nd = vsub[s], g = subb[s];
  for (int c = 0; c < 26; ++c) atomicMaxF(&smax[g * 26 + c], XG[(size_t)nd * 26 + c]);
}

// MLP3 [32->16->10->6] on [pool(26)|goal(6)]; one thread per graph (G=64, tiny).
__global__ void k_mlp3(const float* __restrict__ pool, const float* __restrict__ goal,
                       const float* W0, const float* b0, const float* W1, const float* b1,
                       const float* W2, const float* b2, float* out, int G) {
  int g = blockIdx.x * blockDim.x + threadIdx.x;
  if (g >= G) return;
  float vin[32], h0[16], h1[10];
  for (int k = 0; k < 26; ++k) vin[k] = pool[g * 26 + k];
  for (int k = 0; k < 6; ++k) vin[26 + k] = goal[g * 6 + k];
  for (int j = 0; j < 16; ++j) {
    float s = b0[j];
    for (int k = 0; k < 32; ++k) s += vin[k] * W0[k * 16 + j];
    h0[j] = fmaxf(s, 0.0f);
  }
  for (int j = 0; j < 10; ++j) {
    float s = b1[j];
    for (int k = 0; k < 16; ++k) s += h0[k] * W1[k * 10 + j];
    h1[j] = fmaxf(s, 0.0f);
  }
  for (int j = 0; j < 6; ++j) {
    float s = b2[j];
    for (int k = 0; k < 10; ++k) s += h1[k] * W2[k * 6 + j];
    out[g * 6 + j] = s;
  }
}

// conc = [xm(32) | gemb[b](6) | semb[b](6) | cur[b](7) | goal[b](6)] -> 57
__global__ void k_conc(const float* __restrict__ XM, const float* __restrict__ gemb,
                       const float* __restrict__ semb, const float* __restrict__ cur,
                       const float* __restrict__ goal, const int* __restrict__ batch,
                       float* CONC, int N) {
  int i = blockIdx.x * blockDim.x + threadIdx.x;
  if (i >= N) return;
  int g = batch[i];
  float* o = CONC + (size_t)i * 57;
  for (int k = 0; k < 32; ++k) o[k] = XM[(size_t)i * 32 + k];
  for (int k = 0; k < 6; ++k)  o[32 + k] = gemb[g * 6 + k];
  for (int k = 0; k < 6; ++k)  o[38 + k] = semb[g * 6 + k];
  for (int k = 0; k < 7; ++k)  o[44 + k] = cur[g * 7 + k];
  for (int k = 0; k < 6; ++k)  o[51 + k] = goal[g * 6 + k];
}

__global__ void k_qfinal(const float* __restrict__ QH16, const float* __restrict__ W3,
                         const float* __restrict__ b3, const int* __restrict__ gidx,
                         const int* __restrict__ batch, float* Q, int N) {
  int i = blockIdx.x * blockDim.x + threadIdx.x;
  if (i >= N) return;
  int g = gidx[batch[i]];
  float s = b3[g];
#pragma unroll
  for (int k = 0; k < 16; ++k) s += QH16[(size_t)i * 16 + k] * W3[g * 16 + k];
  Q[i] = s;
}

__global__ void k_mask(const float* __restrict__ Q, const int* __restrict__ amask,
                       float* out, int N) {
  int i = blockIdx.x * blockDim.x + threadIdx.x;
  if (i >= N) return;
  out[i] = (amask[i] == 1) ? Q[i] : -1e8f;
}

// ===========================================================================
extern "C" void kernel_launch(void* const* d_in, const int* in_sizes, int n_in,
                              void* d_out, int out_size, void* d_ws, size_t ws_size,
                              hipStream_t stream) {
  const int N  = in_sizes[0] / 7;
  const int E  = in_sizes[2];
  const int S  = in_sizes[6];
  const int G  = in_sizes[8];
  const int Ef = E + N;
  const int NP = N + 16;   // +16 slack rows for guard-free WMMA A-tile reads

  const float* x         = (const float*)d_in[0];
  const int*   eidx      = (const int*)d_in[1];
  const int*   src       = eidx;
  const int*   dst       = eidx + E;
  const float* edge_attr = (const float*)d_in[2];
  const int*   batch     = (const int*)d_in[3];
  const int*   amask     = (const int*)d_in[4];
  const float* goal      = (const float*)d_in[5];
  const int*   vsub      = (const int*)d_in[6];
  const int*   subb      = (const int*)d_in[7];
  const int*   curnode   = (const int*)d_in[8];
  const float* n0w = (const float*)d_in[9],  *n0b = (const float*)d_in[10], *n0m = (const float*)d_in[11];
  const float* enw = (const float*)d_in[12], *enb = (const float*)d_in[13], *enm = (const float*)d_in[14];

  struct CP { const float *Wl, *bl, *Wr, *br, *We, *att, *bias; int H, C; };
  auto cp = [&](int base, int H, int C) {
    CP p; p.Wl = (const float*)d_in[base]; p.bl = (const float*)d_in[base + 1];
    p.Wr = (const float*)d_in[base + 2]; p.br = (const float*)d_in[base + 3];
    p.We = (const float*)d_in[base + 4]; p.att = (const float*)d_in[base + 5];
    p.bias = (const float*)d_in[base + 6]; p.H = H; p.C = C; return p;
  };
  CP c1 = cp(15, 2, 5), c2 = cp(22, 2, 10), c3 = cp(29, 1, 30), cw = cp(36, 2, 22);
  const float* mW  = (const float*)d_in[43]; const float* mb = (const float*)d_in[44];
  const float* geW0 = (const float*)d_in[45], *geb0 = (const float*)d_in[46];
  const float* geW1 = (const float*)d_in[47], *geb1 = (const float*)d_in[48];
  const float* geW2 = (const float*)d_in[49], *geb2 = (const float*)d_in[50];
  const float* seW0 = (const float*)d_in[51], *seb0 = (const float*)d_in[52];
  const float* seW1 = (const float*)d_in[53], *seb1 = (const float*)d_in[54];
  const float* seW2 = (const float*)d_in[55], *seb2 = (const float*)d_in[56];
  const float* gbW1 = (const float*)d_in[57], *gbb1 = (const float*)d_in[58];
  const float* gbW2 = (const float*)d_in[59], *gbb2 = (const float*)d_in[60];
  const float* qbW1 = (const float*)d_in[61], *qbb1 = (const float*)d_in[62];
  const float* qbW2 = (const float*)d_in[63], *qbb2 = (const float*)d_in[64];
  const float* qbW3 = (const float*)d_in[65], *qbb3 = (const float*)d_in[66];

  // ---------------- workspace carve (floats) ----------------
  float* wsf = (float*)d_ws;
  size_t o = 0;
  auto alloc = [&](size_t n) { float* p = wsf + o; o += n; return p; };
  // zeroed-per-call region first
  float* nodeSum = alloc((size_t)G * 7);
  float* nodeCnt = alloc(G);
  float* edgeSum = alloc(G);
  float* edgeCnt = alloc(G);
  float* nodeVar = alloc((size_t)G * 7);
  float* edgeVar = alloc(G);
  float* einSum = alloc(N);
  float* einCnt = alloc(N);
  float* eoSum  = alloc(N);
  float* eoCnt  = alloc(N);
  size_t zeroBytes = o * sizeof(float);
  // computed regions
  float* nodeMean = alloc((size_t)G * 7);
  float* nodeInv  = alloc((size_t)G * 7);
  float* edgeMean = alloc(G);
  float* edgeInv  = alloc(G);
  float* cur      = alloc((size_t)G * 7);
  float* gmax     = alloc((size_t)G * 26);
  float* smax     = alloc((size_t)G * 26);   // contiguous with gmax
  float* gemb     = alloc((size_t)G * 6);
  float* semb     = alloc((size_t)G * 6);
  int*   gidx     = (int*)alloc(G);
  int*   gstart   = (int*)alloc(G + 1);
  float* EA    = alloc(E);
  float* EOUT  = alloc(N);
  float* H16   = alloc((size_t)NP * 16);
  float* XL    = alloc((size_t)N * 44);
  float* XR    = alloc((size_t)N * 44);
  float* ABUF  = alloc((size_t)Ef * 2);
  float* AMAX  = alloc((size_t)N * 2);
  float* ASUM  = alloc((size_t)N * 2);
  float* H26   = alloc((size_t)NP * 26);
  float* H46   = alloc((size_t)NP * 46);     // contiguous after H26
  float* XBASE = alloc((size_t)NP * 76);
  float* XG    = alloc((size_t)NP * 26);
  float* GH32  = alloc((size_t)NP * 32);     // goal hidden; reused as q hidden32
  float* XM    = alloc((size_t)N * 32);
  // packed (zero-padded) weights: Wp[set][K4][fo16], bp[set][fo16]
  struct PW { float *Wp, *bp; int K4, fo16; };
  auto pwAlloc = [&](int fin, int fo, int nb) {
    PW p; p.K4 = (fin + 3) & ~3; p.fo16 = (fo + 15) & ~15;
    p.Wp = alloc((size_t)nb * p.K4 * p.fo16);
    p.bp = alloc((size_t)nb * p.fo16);
    return p;
  };
  PW pc1l = pwAlloc(16, 10, 1),  pc1r = pwAlloc(16, 10, 1);
  PW pc2l = pwAlloc(26, 20, 1),  pc2r = pwAlloc(26, 20, 1);
  PW pc3l = pwAlloc(46, 30, 1),  pc3r = pwAlloc(46, 30, 1);
  PW pcwl = pwAlloc(57, 44, 1),  pcwr = pwAlloc(57, 44, 1);
  PW pmrg = pwAlloc(102, 32, 1);
  PW pgb1 = pwAlloc(76, 32, 6),  pgb2 = pwAlloc(32, 26, 6);
  PW pqb1 = pwAlloc(44, 32, 6),  pqb2 = pwAlloc(32, 16, 6);
  // lifetime-based aliases (regions free by the time these are written)
  float* CONC  = XBASE;   // 57N <= 76N; XBASE dead after merge
  float* CWOUT = H26;     // 44*NP <= (26+46)*NP; H26/H46 dead after conv3 pack
  float* QH16  = H16;     // H16 dead after conv1
  float* Q     = ABUF;    // ABUF dead after conv_whole edge passes
  (void)ws_size; (void)n_in; (void)out_size;

  dim3 blk(TPB);
  dim3 gN((N + TPB - 1) / TPB), gEe((E + TPB - 1) / TPB), gEf((Ef + TPB - 1) / TPB);
  dim3 gG((G + 63) / 64), blkG(64);

  // ---------------- weight packing (independent; runs first) ----------------
  auto packW = [&](const PW& p, const float* W, const float* b, int fin, int fo, int nb) {
    int tot = nb * p.K4 * p.fo16;
    k_packW<<<dim3((tot + TPB - 1) / TPB), blk, 0, stream>>>(W, b, p.Wp, p.bp,
                                                             fin, fo, p.K4, p.fo16, nb);
  };
  packW(pc1l, c1.Wl, c1.bl, 16, 10, 1);  packW(pc1r, c1.Wr, c1.br, 16, 10, 1);
  packW(pc2l, c2.Wl, c2.bl, 26, 20, 1);  packW(pc2r, c2.Wr, c2.br, 26, 20, 1);
  packW(pc3l, c3.Wl, c3.bl, 46, 30, 1);  packW(pc3r, c3.Wr, c3.br, 46, 30, 1);
  packW(pcwl, cw.Wl, cw.bl, 57, 44, 1);  packW(pcwr, cw.Wr, cw.br, 57, 44, 1);
  packW(pmrg, mW, mb, 102, 32, 1);
  packW(pgb1, gbW1, gbb1, 76, 32, 6);    packW(pgb2, gbW2, gbb2, 32, 26, 6);
  packW(pqb1, qbW1, qbb1, 44, 32, 6);    packW(pqb2, qbW2, qbb2, 32, 16, 6);

  // ---------------- GraphNorm + feature assembly ----------------
  hipMemsetAsync(wsf, 0, zeroBytes, stream);
  k_node_stats1<<<gN, blk, 0, stream>>>(x, batch, nodeSum, nodeCnt, N);
  k_edge_stats1<<<gEe, blk, 0, stream>>>(edge_attr, src, batch, edgeSum, edgeCnt, E);
  k_means<<<gG, blkG, 0, stream>>>(nodeSum, nodeCnt, edgeSum, edgeCnt, nodeMean, edgeMean, G);
  k_node_stats2<<<gN, blk, 0, stream>>>(x, batch, nodeMean, n0m, nodeVar, N);
  k_edge_stats2<<<gEe, blk, 0, stream>>>(edge_attr, src, batch, edgeMean, enm, edgeVar, E);
  k_inv<<<gG, blkG, 0, stream>>>(nodeVar, nodeCnt, edgeVar, edgeCnt, nodeInv, edgeInv, G);
  k_ea<<<gEe, blk, 0, stream>>>(edge_attr, src, dst, batch, edgeMean, edgeInv,
                                enw, enb, enm, EA, einSum, einCnt, eoSum, eoCnt, E);
  k_h16a<<<gN, blk, 0, stream>>>(x, batch, nodeMean, nodeInv, n0w, n0b, n0m,
                                 einSum, einCnt, eoSum, eoCnt, H16, EOUT, N);
  k_cur_gidx<<<gG, blkG, 0, stream>>>(H16, curnode, goal, cur, gidx, G);
  k_ranges<<<gN, blk, 0, stream>>>(batch, gstart, N, G);
  k_h16b<<<gN, blk, 0, stream>>>(cur, batch, H16, N);

  // ---------------- generic GATv2 conv ----------------
  auto conv = [&](const float* in, int si, const CP& p, const PW& pl, const PW& pr,
                  float* dest, int so, int identLen) {
    int HC = p.H * p.C;
    int rowT = (N + 15) / 16, colT = pl.fo16 >> 4;
    k_gemm<0><<<dim3(rowT * colT), dim3(32), 0, stream>>>(
        in, si, pl.K4, nullptr, 0, pl.K4, pl.Wp, pl.bp, XL, HC, HC, pl.fo16, N, 0);
    k_gemm<0><<<dim3(rowT * colT), dim3(32), 0, stream>>>(
        in, si, pr.K4, nullptr, 0, pr.K4, pr.Wp, pr.bp, XR, HC, HC, pr.fo16, N, 0);
    k_fill_att<<<dim3((N * p.H + TPB - 1) / TPB), blk, 0, stream>>>(AMAX, ASUM, N * p.H);
    hipMemsetAsync(dest, 0, (size_t)N * so * sizeof(float), stream);
    k_edge1<<<gEf, blk, 0, stream>>>(src, dst, EA, EOUT, XL, XR, p.We, p.att,
                                     ABUF, AMAX, p.H, p.C, E, Ef);
    k_edge2<<<gEf, blk, 0, stream>>>(dst, ABUF, AMAX, ASUM, p.H, E, Ef);
    k_edge3<<<gEf, blk, 0, stream>>>(src, dst, ABUF, ASUM, XL, dest, so, p.H, p.C, E, Ef);
    k_pack<<<gN, blk, 0, stream>>>(dest, so, HC, p.bias, in, si, identLen, N);
  };

  conv(H16, 16, c1, pc1l, pc1r, H26, 26, 16);     // -> [N,26]
  conv(H26, 26, c2, pc2l, pc2r, H46, 46, 26);     // -> [N,46]
  conv(H46, 46, c3, pc3l, pc3r, XBASE, 76, 46);   // -> [N,76]

  // ---------------- goal branch (per-graph WMMA) ----------------
  int maxg = (N + G - 1) / G + 1;
  int TPG = (maxg + 15) / 16 + 1;
  auto branch = [&](const float* in, int si, const PW& p, float* outp, int so, int fo,
                    int relu) {
    int colT = p.fo16 >> 4;
    k_gemm_branch<<<dim3(G * colT * TPG), dim3(32), 0, stream>>>(
        in, si, p.K4, p.Wp, p.bp, outp, so, fo, p.fo16, gstart, gidx, TPG, relu);
  };
  branch(XBASE, 76, pgb1, GH32, 32, 32, 1);
  branch(GH32, 32, pgb2, XG, 26, 26, 0);

  // ---------------- pools + embedder MLPs ----------------
  k_fill_neg<<<dim3((G * 52 + TPB - 1) / TPB), blk, 0, stream>>>(gmax, G * 52);
  k_gmax<<<gN, blk, 0, stream>>>(XG, batch, gmax, N);
  k_smax<<<dim3((S + TPB - 1) / TPB), blk, 0, stream>>>(XG, vsub, subb, smax, S);
  k_mlp3<<<gG, blkG, 0, stream>>>(gmax, goal, geW0, geb0, geW1, geb1, geW2, geb2, gemb, G);
  k_mlp3<<<gG, blkG, 0, stream>>>(smax, goal, seW0, seb0, seW1, seb1, seW2, seb2, semb, G);

  // ---------------- merge (split-input WMMA GEMM over [XBASE|XG]) ----------
  {
    int rowT = (N + 15) / 16, colT = pmrg.fo16 >> 4;
    k_gemm<1><<<dim3(rowT * colT), dim3(32), 0, stream>>>(
        XBASE, 76, 76, XG, 26, pmrg.K4, pmrg.Wp, pmrg.bp, XM, 32, 32, pmrg.fo16, N, 0);
  }

  // ---------------- conc + conv_whole ----------------
  k_conc<<<gN, blk, 0, stream>>>(XM, gemb, semb, cur, goal, batch, CONC, N);
  conv(CONC, 57, cw, pcwl, pcwr, CWOUT, 44, 0);   // -> [N,44], no ident concat

  // ---------------- qvalue branch ----------------
  branch(CWOUT, 44, pqb1, GH32, 32, 32, 1);
  branch(GH32, 32, pqb2, QH16, 16, 16, 1);
  k_qfinal<<<gN, blk, 0, stream>>>(QH16, qbW3, qbb3, gidx, batch, Q, N);
  k_mask<<<gN, blk, 0, stream>>>(Q, amask, (float*)d_out, N);
}